// KW_CascadedBranch_45861660787528
// MI455X (gfx1250) — compile-verified
//
#include <hip/hip_runtime.h>
#include <math.h>

// ---------------------------------------------------------------------------
// Problem constants (from reference)
// ---------------------------------------------------------------------------
#define M_ROWS   1024      // BSZ(128) * KW(8)
#define D_MODEL  768
#define TEXT_DIM 512
#define VOCAB    49408     // % 16 == 0 (3088 tiles), % 4 == 0 (12352 float4/row)
#define VOCAB4   (VOCAB / 4)
#define EPS      1e-8f

typedef __attribute__((ext_vector_type(2))) float v2f;
typedef __attribute__((ext_vector_type(4))) float v4f;
typedef __attribute__((ext_vector_type(8))) float v8f;

// ---------------------------------------------------------------------------
// WMMA helper: D = A(16x4) * B(4x16) + C, f32, one tile per wave32.
// A per-lane: lanes 0-15 -> row M=lane, K=0..1 ; lanes 16-31 -> K=2..3
// B per-lane: VGPR0: lanes 0-15 K=0, lanes 16-31 K=2 ; VGPR1: K=1 / K=3
// C/D: VGPR i: lanes 0-15 -> (M=i, N=lane) ; lanes 16-31 -> (M=8+i, N=lane-16)
// ---------------------------------------------------------------------------
__device__ __forceinline__ v8f wmma_f32_4(v2f a, v2f b, v8f c) {
    return __builtin_amdgcn_wmma_f32_16x16x4_f32(
        /*neg_a=*/false, a, /*neg_b=*/false, b,
        /*c_mod=*/(short)0, c, /*reuse_a=*/false, /*reuse_b=*/false);
}

// ---------------------------------------------------------------------------
// 1) kw = keywords @ W_proj + b_proj      [1024 x 768] x [768 x 512]
//    grid (64, 32), block 32 (one wave per 16x16 tile), 192 WMMAs per tile
// ---------------------------------------------------------------------------
__global__ void proj_kernel(const float* __restrict__ keywords,
                            const float* __restrict__ Wp,
                            const float* __restrict__ bp,
                            float* __restrict__ kw) {
    const int lane  = threadIdx.x;
    const int m0    = blockIdx.x << 4;
    const int n0    = blockIdx.y << 4;
    const int l15   = lane & 15;
    const int khalf = (lane >> 4) << 1;   // 0 or 2

    const float* arow = keywords + (size_t)(m0 + l15) * D_MODEL + khalf;
    const float* bcol = Wp + (size_t)khalf * TEXT_DIM + n0 + l15;

    v8f c = {};
    for (int k = 0; k < D_MODEL; k += 4) {
        v2f a = *(const v2f*)(arow + k);
        v2f b;
        b.x = bcol[(size_t)(k)     * TEXT_DIM];
        b.y = bcol[(size_t)(k + 1) * TEXT_DIM];
        c = wmma_f32_4(a, b, c);
    }
    const float bias  = bp[n0 + l15];
    const int   col   = n0 + l15;
    const int   rbase = m0 + ((lane >> 4) << 3);
#pragma unroll
    for (int i = 0; i < 8; ++i)
        kw[(size_t)(rbase + i) * TEXT_DIM + col] = c[i] + bias;
}

// ---------------------------------------------------------------------------
// 2a) Normalize kw rows in place (1024 rows of 512).  One wave per row.
// ---------------------------------------------------------------------------
__global__ void kwnorm_kernel(float* __restrict__ kw) {
    const int wave = threadIdx.x >> 5;
    const int lane = threadIdx.x & 31;
    const int row  = (blockIdx.x << 3) + wave;
    float* p = kw + (size_t)row * TEXT_DIM + lane * 16;

    v4f v[4];
    float s = 0.f;
#pragma unroll
    for (int i = 0; i < 4; ++i) {
        v[i] = *(const v4f*)(p + i * 4);
        s += v[i].x * v[i].x + v[i].y * v[i].y + v[i].z * v[i].z + v[i].w * v[i].w;
    }
#pragma unroll
    for (int off = 16; off; off >>= 1) s += __shfl_xor(s, off, 32);
    const float inv = 1.0f / fmaxf(sqrtf(s), EPS);
#pragma unroll
    for (int i = 0; i < 4; ++i) {
        v4f o = { v[i].x * inv, v[i].y * inv, v[i].z * inv, v[i].w * inv };
        *(v4f*)(p + i * 4) = o;
    }
}

// ---------------------------------------------------------------------------
// 2b) inv_emb[v] = 1/max(||W_emb[v]||, EPS)   (49408 rows). One wave per row.
// ---------------------------------------------------------------------------
__global__ void embnorm_kernel(const float* __restrict__ Wemb,
                               float* __restrict__ inv_emb) {
    const int wave = threadIdx.x >> 5;
    const int lane = threadIdx.x & 31;
    const int row  = (blockIdx.x << 3) + wave;
    const float* p = Wemb + (size_t)row * TEXT_DIM + lane * 16;

    float s = 0.f;
#pragma unroll
    for (int i = 0; i < 4; ++i) {
        v4f v = *(const v4f*)(p + i * 4);
        s += v.x * v.x + v.y * v.y + v.z * v.z + v.w * v.w;
    }
#pragma unroll
    for (int off = 16; off; off >>= 1) s += __shfl_xor(s, off, 32);
    if (lane == 0) inv_emb[row] = 1.0f / fmaxf(sqrtf(s), EPS);
}

// ---------------------------------------------------------------------------
// 3) cos_score = kw_n @ W_emb^T, columns scaled by inv_emb in the epilogue.
//    grid (64, 3088), block 32.  128 WMMAs per tile.
//    W_emb (101 MB) + kw_n (2 MB) are L2-resident -> operand reads stay in L2.
// ---------------------------------------------------------------------------
__global__ void cos_kernel(const float* __restrict__ kwn,
                           const float* __restrict__ Wemb,
                           const float* __restrict__ inv_emb,
                           float* __restrict__ cos_out) {
    const int lane  = threadIdx.x;
    const int m0    = blockIdx.x << 4;
    const int v0    = blockIdx.y << 4;
    const int l15   = lane & 15;
    const int khalf = (lane >> 4) << 1;

    const float* arow = kwn  + (size_t)(m0 + l15) * TEXT_DIM + khalf;
    const float* brow = Wemb + (size_t)(v0 + l15) * TEXT_DIM + khalf;

    v8f c = {};
    for (int k = 0; k < TEXT_DIM; k += 4) {
        v2f a = *(const v2f*)(arow + k);
        v2f b = *(const v2f*)(brow + k);   // B[k][n] = W_emb[v0+n][k0+k]
        c = wmma_f32_4(a, b, c);
    }
    const float scl   = inv_emb[v0 + l15];
    const int   col   = v0 + l15;
    const int   rbase = m0 + ((lane >> 4) << 3);
#pragma unroll
    for (int i = 0; i < 8; ++i)
        cos_out[(size_t)(rbase + i) * VOCAB + col] = c[i] * scl;
}

// ---------------------------------------------------------------------------
// 4) Online softmax stats: one 256-thread workgroup per row, single read of
//    the row in float4 (b128) chunks.
// ---------------------------------------------------------------------------
__global__ void softmax_stats_kernel(const float* __restrict__ cosb,
                                     float* __restrict__ rowmax,
                                     float* __restrict__ rowinv) {
    __shared__ float sm[8], ss[8];
    const int row = blockIdx.x;
    const int tid = threadIdx.x;
    const v4f* r4 = (const v4f*)(cosb + (size_t)row * VOCAB);

    float m = -INFINITY, s = 0.f;
    for (int v4 = tid; v4 < VOCAB4; v4 += 256) {
        const v4f x = r4[v4];
        const float xs[4] = { x.x, x.y, x.z, x.w };
#pragma unroll
        for (int j = 0; j < 4; ++j) {
            if (xs[j] > m) { s = s * __expf(m - xs[j]) + 1.0f; m = xs[j]; }
            else           { s += __expf(xs[j] - m); }
        }
    }
#pragma unroll
    for (int off = 16; off; off >>= 1) {
        const float m2 = __shfl_xor(m, off, 32);
        const float s2 = __shfl_xor(s, off, 32);
        const float mm = fmaxf(m, m2);
        s = s * __expf(m - mm) + s2 * __expf(m2 - mm);
        m = mm;
    }
    if ((tid & 31) == 0) { sm[tid >> 5] = m; ss[tid >> 5] = s; }
    __syncthreads();
    if (tid == 0) {
        float M = sm[0], S = ss[0];
#pragma unroll
        for (int i = 1; i < 8; ++i) {
            const float mm = fmaxf(M, sm[i]);
            S = S * __expf(M - mm) + ss[i] * __expf(sm[i] - mm);
            M = mm;
        }
        rowmax[row] = M;
        rowinv[row] = 1.0f / S;
    }
}

// ---------------------------------------------------------------------------
// 5) subword_prob = exp(cos - M) * invS.  Bandwidth-critical 400 MB stream:
//    b128 loads/stores, one float4 per thread.
//    Total float4s = 1024*49408/4 = 12,648,448 = 49408 blocks x 256 threads.
// ---------------------------------------------------------------------------
__global__ void prob_kernel(const float* __restrict__ cosb,
                            const float* __restrict__ rowmax,
                            const float* __restrict__ rowinv,
                            float* __restrict__ prob) {
    const size_t idx4 = (size_t)blockIdx.x * 256 + threadIdx.x;
    const int    row  = (int)(idx4 / VOCAB4);      // VOCAB % 4 == 0: no row split
    const float  M    = rowmax[row];
    const float  invS = rowinv[row];

    const v4f x = ((const v4f*)cosb)[idx4];
    v4f p;
    p.x = __expf(x.x - M) * invS;
    p.y = __expf(x.y - M) * invS;
    p.z = __expf(x.z - M) * invS;
    p.w = __expf(x.w - M) * invS;
    ((v4f*)prob)[idx4] = p;
}

// ---------------------------------------------------------------------------
// 6) kw_out = subword_prob @ W_emb   [1024 x 49408] x [49408 x 512]
//    grid (64, 32), block 32.  12352 WMMAs per tile.
// ---------------------------------------------------------------------------
__global__ void out_gemm_kernel(const float* __restrict__ prob,
                                const float* __restrict__ Wemb,
                                float* __restrict__ out) {
    const int lane  = threadIdx.x;
    const int m0    = blockIdx.x << 4;
    const int d0    = blockIdx.y << 4;
    const int l15   = lane & 15;
    const int khalf = (lane >> 4) << 1;

    const float* arow = prob + (size_t)(m0 + l15) * VOCAB + khalf;
    const float* bcol = Wemb + (size_t)khalf * TEXT_DIM + d0 + l15;

    v8f c = {};
#pragma unroll 4
    for (int k = 0; k < VOCAB; k += 4) {
        v2f a = *(const v2f*)(arow + k);
        v2f b;
        b.x = bcol[(size_t)(k)     * TEXT_DIM];
        b.y = bcol[(size_t)(k + 1) * TEXT_DIM];
        c = wmma_f32_4(a, b, c);
    }
    const int col   = d0 + l15;
    const int rbase = m0 + ((lane >> 4) << 3);
#pragma unroll
    for (int i = 0; i < 8; ++i)
        out[(size_t)(rbase + i) * TEXT_DIM + col] = c[i];
}

// ---------------------------------------------------------------------------
// Launcher
// ---------------------------------------------------------------------------
extern "C" void kernel_launch(void* const* d_in, const int* in_sizes, int n_in,
                              void* d_out, int out_size, void* d_ws, size_t ws_size,
                              hipStream_t stream) {
    const float* keywords = (const float*)d_in[0];   // [1024, 768]
    const float* W_proj   = (const float*)d_in[1];   // [768, 512]
    const float* b_proj   = (const float*)d_in[2];   // [512]
    const float* W_emb    = (const float*)d_in[3];   // [49408, 512]

    // Output layout (return order): kw_out | subword_prob | cos_score
    float* out_f    = (float*)d_out;
    float* kw_out   = out_f;                                        // 1024*512
    float* prob_out = out_f + (size_t)M_ROWS * TEXT_DIM;            // 1024*49408
    float* cos_out  = prob_out + (size_t)M_ROWS * VOCAB;            // 1024*49408

    // Scratch layout (~2.3 MB)
    float* w       = (float*)d_ws;
    float* kw_n    = w;                                             // 1024*512
    float* inv_emb = kw_n + (size_t)M_ROWS * TEXT_DIM;              // 49408
    float* rowmax  = inv_emb + VOCAB;                               // 1024
    float* rowinv  = rowmax + M_ROWS;                               // 1024

    // 1) projection + bias
    proj_kernel<<<dim3(M_ROWS / 16, TEXT_DIM / 16), 32, 0, stream>>>(
        keywords, W_proj, b_proj, kw_n);
    // 2) row norms
    kwnorm_kernel<<<M_ROWS / 8, 256, 0, stream>>>(kw_n);
    embnorm_kernel<<<VOCAB / 8, 256, 0, stream>>>(W_emb, inv_emb);
    // 3) cosine GEMM (epilogue column scale by inv_emb)
    cos_kernel<<<dim3(M_ROWS / 16, VOCAB / 16), 32, 0, stream>>>(
        kw_n, W_emb, inv_emb, cos_out);
    // 4) online softmax stats (single b128 read of cos_score)
    softmax_stats_kernel<<<M_ROWS, 256, 0, stream>>>(cos_out, rowmax, rowinv);
    // 5) probability write (bandwidth-critical b128 stream)
    prob_kernel<<<VOCAB, 256, 0, stream>>>(cos_out, rowmax, rowinv, prob_out);
    // 6) soft assignment GEMM
    out_gemm_kernel<<<dim3(M_ROWS / 16, TEXT_DIM / 16), 32, 0, stream>>>(
        prob_out, W_emb, kw_out);
}